// QuarCenterTripletLoss_36953898615127
// MI455X (gfx1250) — compile-verified
//
#include <hip/hip_runtime.h>
#include <math.h>
#include <stdint.h>

typedef __attribute__((ext_vector_type(2))) float v2f;
typedef __attribute__((ext_vector_type(8))) float v8f;

#define N_ROWS 4096
#define N_DIM  512
#define N_CLS  512
#define HALF_N 2048
#define MARGIN_F 0.3f

// Workspace layout (float offsets)
#define WS_CENTERS 0                       // 512*512 class centers (S_all/cnt, 0 if empty)
#define WS_CNT    (N_CLS * N_DIM)          // 512: class counts (float)
#define WS_PC     (WS_CNT + N_CLS)         // 512: |center_c|^2
#define WS_PIN    (WS_PC + N_CLS)          // 4096: |x_i|^2
#define WS_SUMPC  (WS_PIN + N_ROWS)        // scalar: sum of dist_pc
#define WS_SUMAN  (WS_SUMPC + 1)           // scalar: sum of dist_an

// ---------------- reduction helpers ----------------
__device__ __forceinline__ float blockReduce256(float v, float* red) {
    const int lane = threadIdx.x & 31;
    const int wid  = threadIdx.x >> 5;
#pragma unroll
    for (int o = 16; o; o >>= 1) v += __shfl_down(v, o, 32);
    if (lane == 0) red[wid] = v;
    __syncthreads();
    float r = 0.0f;
    if (threadIdx.x < 8) r = red[threadIdx.x];
    if (wid == 0) {
#pragma unroll
        for (int o = 4; o; o >>= 1) r += __shfl_down(r, o, 32);
    }
    __syncthreads();
    return r;  // valid in thread 0
}

// ---------------- kernel 0: zero accumulators ----------------
__global__ void init_kernel(float* ws) {
    if (threadIdx.x == 0) {
        ws[WS_SUMPC] = 0.0f;
        ws[WS_SUMAN] = 0.0f;
    }
}

// ---------------- kernel 1: per-class sums, centers, |C|^2, dist_pc ----------------
// One block (256 threads) per class c. Each thread owns 2 dims.
__global__ __launch_bounds__(256)
void class_kernel(const float* __restrict__ x, const int* __restrict__ t,
                  float* __restrict__ ws) {
    const int c   = blockIdx.x;
    const int tid = threadIdx.x;
    const int d0  = tid * 2;

    __shared__ int   lab[N_ROWS];
    __shared__ float red[8];

    for (int j = tid; j < N_ROWS; j += 256) lab[j] = t[j];
    __syncthreads();

    float sA0 = 0.f, sA1 = 0.f;  // sum over all rows of class c
    float sI0 = 0.f, sI1 = 0.f;  // sum over IR rows (j>=2048) whose own label == c
    float sR0 = 0.f, sR1 = 0.f;  // sum over RGB rows j-2048 whose PAIRED IR label == c
    int cntAll = 0, cntIND = 0;

    for (int j = 0; j < N_ROWS; ++j) {
        if (lab[j] == c) {
            const float a0 = x[j * N_DIM + d0];
            const float a1 = x[j * N_DIM + d0 + 1];
            sA0 += a0; sA1 += a1; cntAll++;
            if (j >= HALF_N) {
                sI0 += a0; sI1 += a1;
                sR0 += x[(j - HALF_N) * N_DIM + d0];
                sR1 += x[(j - HALF_N) * N_DIM + d0 + 1];
                cntIND++;
            }
        }
    }

    // global class center (0 for empty class) + |center|^2 + count
    const float invA = (cntAll > 0) ? (1.0f / (float)cntAll) : 0.0f;
    const float c0 = sA0 * invA, c1 = sA1 * invA;
    ws[WS_CENTERS + c * N_DIM + d0]     = c0;
    ws[WS_CENTERS + c * N_DIM + d0 + 1] = c1;
    const float pcsum = blockReduce256(c0 * c0 + c1 * c1, red);
    if (tid == 0) {
        ws[WS_PC  + c] = pcsum;
        ws[WS_CNT + c] = (float)cntAll;
    }

    // dist_pc: rows i<2048 use centerIR[targets[2048+i]]; rows j>=2048 use centerRGBD[targets[j]]
    float acc_pc = 0.0f;  // thread 0 only
    if (cntIND > 0) {
        const float inv = 1.0f / (float)cntIND;
        const float i0 = sI0 * inv, i1 = sI1 * inv;  // centerIR[c]
        const float r0 = sR0 * inv, r1 = sR1 * inv;  // centerRGBD_byIND[c]
        for (int j = HALF_N; j < N_ROWS; ++j) {
            if (lab[j] == c) {
                const float e0 = x[j * N_DIM + d0]     - r0;
                const float e1 = x[j * N_DIM + d0 + 1] - r1;
                const float f0 = x[(j - HALF_N) * N_DIM + d0]     - i0;
                const float f1 = x[(j - HALF_N) * N_DIM + d0 + 1] - i1;
                const float sIRrow  = blockReduce256(e0 * e0 + e1 * e1, red);
                const float sRGBrow = blockReduce256(f0 * f0 + f1 * f1, red);
                if (tid == 0) acc_pc += sqrtf(sIRrow) + sqrtf(sRGBrow);
            }
        }
    }
    if (tid == 0) atomicAdd(&ws[WS_SUMPC], acc_pc);
}

// ---------------- kernel 2: |x_i|^2 ----------------
__global__ __launch_bounds__(128)
void pin_kernel(const float* __restrict__ x, float* __restrict__ ws) {
    const int row = blockIdx.x;
    const int tid = threadIdx.x;
    __shared__ float red[4];
    float v = 0.0f;
    for (int d = tid; d < N_DIM; d += 128) {
        const float a = x[row * N_DIM + d];
        v += a * a;
    }
#pragma unroll
    for (int o = 16; o; o >>= 1) v += __shfl_down(v, o, 32);
    if ((tid & 31) == 0) red[tid >> 5] = v;
    __syncthreads();
    if (tid == 0) ws[WS_PIN + row] = red[0] + red[1] + red[2] + red[3];
}

// ---------------- kernel 3: WMMA GEMM (x @ centers^T) + hinge reduction ----------------
// 8 waves/block; each wave owns a 16-row A strip. All waves share the 16-class B
// tile, staged into double-buffered LDS (2x32KB) with GLOBAL_LOAD_ASYNC_TO_LDS_B128
// (ASYNCcnt), prefetching tile i+1 while computing tile i. B-fragments come from
// LDS (ds_load_b64); A-fragments from global (L2-resident).
// Fragments (V_WMMA_F32_16X16X4_F32):
//   A 16x4: lane l -> m = l&15, k = k0 + 2*(l>>4) + {0,1}
//   B 4x16: lane l -> n = l&15, same k pair; B[k][n] = centers[n][k]
//   C/D:    lane l, reg r -> m = r + 8*(l>>4), n = l&15
#define B_TILE_FLOATS (16 * N_DIM)       // 8192 floats = 32KB
#define B_TILE_CHUNKS (B_TILE_FLOATS / 4)  // 2048 16-byte chunks

__global__ __launch_bounds__(256)
void an_kernel(const float* __restrict__ x, const int* __restrict__ t,
               float* __restrict__ ws) {
    const float* __restrict__ centers = ws + WS_CENTERS;
    const float* __restrict__ pc      = ws + WS_PC;
    const float* __restrict__ pin     = ws + WS_PIN;
    const float* __restrict__ cntf    = ws + WS_CNT;

    __shared__ float sB[2][B_TILE_FLOATS];   // 64KB double buffer

    const int wave = (int)((blockIdx.x * blockDim.x + threadIdx.x) >> 5);  // 0..255
    const int lane = (int)(threadIdx.x & 31);
    const int m0   = wave * 16;
    const int hi   = lane >> 4;
    const int ln   = lane & 15;

    const float* arow = x + (size_t)(m0 + ln) * N_DIM + 2 * hi;

    // hoist row-invariant data
    float pin_r[8];
    int   t_r[8];
#pragma unroll
    for (int r = 0; r < 8; ++r) {
        const int m = m0 + r + 8 * hi;
        pin_r[r] = pin[m];
        t_r[r]   = t[m];
    }

    float rs[8];
#pragma unroll
    for (int r = 0; r < 8; ++r) rs[r] = 0.0f;

    // block-cooperative async stage of 16 center rows into sB[buf]
    auto stage = [&](int tile, int buf) {
        const float* src = centers + (size_t)tile * 16 * N_DIM;
#pragma unroll
        for (int it = 0; it < B_TILE_CHUNKS / 256; ++it) {
            const int idx = it * 256 + (int)threadIdx.x;            // 16B chunk id
            const unsigned lds_off = (unsigned)(uintptr_t)&sB[buf][idx * 4];
            const uint64_t gaddr   = (uint64_t)(uintptr_t)(src + idx * 4);
            asm volatile("global_load_async_to_lds_b128 %0, %1, off"
                         :: "v"(lds_off), "v"(gaddr) : "memory");
        }
    };

    // prologue: load tile 0
    stage(0, 0);
    asm volatile("s_wait_asynccnt 0x0" ::: "memory");
    __syncthreads();

    const int NTILES = N_CLS / 16;  // 32
    for (int i = 0; i < NTILES; ++i) {
        const int buf = i & 1;
        // prefetch next tile into the other buffer (safe: it was last read in
        // iteration i-1, whose end-of-iteration barrier already passed)
        if (i + 1 < NTILES) stage(i + 1, buf ^ 1);

        // ---- compute 128 fp32 WMMAs from LDS-resident B tile ----
        const float* brow = &sB[buf][ln * N_DIM + 2 * hi];
        v8f acc = {0.f, 0.f, 0.f, 0.f, 0.f, 0.f, 0.f, 0.f};
        for (int k0 = 0; k0 < N_DIM; k0 += 4) {
            const v2f a = *(const v2f*)(arow + k0);
            const v2f b = *(const v2f*)(brow + k0);
            acc = __builtin_amdgcn_wmma_f32_16x16x4_f32(
                /*neg_a=*/false, a, /*neg_b=*/false, b,
                /*c_mod=*/(short)0, acc, /*reuse_a=*/false, /*reuse_b=*/false);
        }

        // ---- hinge epilogue for this 16-class tile ----
        const int   n   = i * 16 + ln;
        const float pcn = pc[n];
        const float w   = cntf[n];
#pragma unroll
        for (int r = 0; r < 8; ++r) {
            float d2 = pin_r[r] + pcn - 2.0f * acc[r];
            d2 = fmaxf(d2, 1e-12f);
            const float dist = sqrtf(d2);
            float h = MARGIN_F - dist;
            h = (h > 0.0f) ? h : 0.0f;
            rs[r] += (t_r[r] == n) ? 0.0f : (w * h);
        }

        if (i + 1 < NTILES) {
            // my async loads for tile i+1 done; barrier => everyone's are done,
            // and everyone is finished reading sB[buf] for tile i.
            asm volatile("s_wait_asynccnt 0x0" ::: "memory");
            __syncthreads();
        }
    }

    // reduce over the 16 lanes sharing each row (xor masks stay within each half)
#pragma unroll
    for (int r = 0; r < 8; ++r) {
        float v = rs[r];
#pragma unroll
        for (int o = 8; o; o >>= 1) v += __shfl_xor(v, o, 32);
        rs[r] = v;
    }
    if (ln == 0) {
#pragma unroll
        for (int r = 0; r < 8; ++r) {
            const float negcnt = (float)N_ROWS - cntf[t_r[r]];
            atomicAdd(&ws[WS_SUMAN], rs[r] / negcnt);
        }
    }
}

// ---------------- kernel 4: final scalar ----------------
__global__ void final_kernel(const float* __restrict__ ws, float* __restrict__ out) {
    const float inv_n = 1.0f / (float)N_ROWS;
    out[0] = 0.5f * (ws[WS_SUMPC] * inv_n + ws[WS_SUMAN] * inv_n);
}

// ---------------- launch ----------------
extern "C" void kernel_launch(void* const* d_in, const int* in_sizes, int n_in,
                              void* d_out, int out_size, void* d_ws, size_t ws_size,
                              hipStream_t stream) {
    (void)in_sizes; (void)n_in; (void)out_size; (void)ws_size;
    const float* x = (const float*)d_in[0];
    const int*   t = (const int*)d_in[1];
    float* ws  = (float*)d_ws;
    float* out = (float*)d_out;

    init_kernel<<<1, 64, 0, stream>>>(ws);
    class_kernel<<<N_CLS, 256, 0, stream>>>(x, t, ws);
    pin_kernel<<<N_ROWS, 128, 0, stream>>>(x, ws);
    an_kernel<<<32, 256, 0, stream>>>(x, t, ws);   // 256 waves, one 16-row strip each
    final_kernel<<<1, 1, 0, stream>>>(ws, out);
}